// GraphEncoder_69441031242027
// MI455X (gfx1250) — compile-verified
//
#include <hip/hip_runtime.h>

// ---------------------------------------------------------------------------
// 3-layer GCN forward on gfx1250 (MI455X).
//   per layer:  h = (x * norm_src[:,None]) @ W      (WMMA f32 16x16x4, LDS-staged W)
//               agg[dst] += h[src]                  (f32 global atomics, float4 gather)
//               x = relu(agg * norm_dst[:,None] + b)
//   output:     mean(x, axis=0)  -> 64 floats
// ---------------------------------------------------------------------------

typedef float v2f __attribute__((ext_vector_type(2)));
typedef float v4f __attribute__((ext_vector_type(4)));
typedef float v8f __attribute__((ext_vector_type(8)));

#define HID 64
#define IN_DIM 23

// ------------------------------- utilities --------------------------------

__global__ void fill_zero_f32(float* __restrict__ p, long long n) {
    long long i = (long long)blockIdx.x * blockDim.x + threadIdx.x;
    if (i < n) p[i] = 0.0f;
}

__global__ void degree_accum(const int* __restrict__ src, const int* __restrict__ dst,
                             float* __restrict__ deg_out, float* __restrict__ deg_in,
                             int n_edges) {
    int e = blockIdx.x * blockDim.x + threadIdx.x;
    if (e < n_edges) {
        atomicAdd(&deg_out[src[e]], 1.0f);
        atomicAdd(&deg_in[dst[e]], 1.0f);
    }
}

__global__ void norm_finalize(float* __restrict__ a, float* __restrict__ b, int n) {
    int i = blockIdx.x * blockDim.x + threadIdx.x;
    if (i < n) {
        a[i] = rsqrtf(fmaxf(a[i], 1.0f));
        b[i] = rsqrtf(fmaxf(b[i], 1.0f));
    }
}

// xs[r, k] = (k < K) ? x[r, k] * norm[r] : 0     (xs is [n_rows, KP], KP%4==0)
__global__ void prescale_pad(const float* __restrict__ X, int ldx, int K,
                             const float* __restrict__ norm,
                             float* __restrict__ XS, int KP, int n_rows) {
    long long t = (long long)blockIdx.x * blockDim.x + threadIdx.x;
    if (t >= (long long)n_rows * KP) return;
    const int r = (int)(t / KP);
    const int k = (int)(t - (long long)r * KP);
    XS[t] = (k < K) ? X[(long long)r * ldx + k] * norm[r] : 0.0f;
}

// Wp[k, c] = (k < K) ? W[k, c] : 0               (Wp is [KP, 64])
__global__ void pad_weight(const float* __restrict__ W, int K,
                           float* __restrict__ Wp, int KP) {
    int t = blockIdx.x * blockDim.x + threadIdx.x;
    if (t >= KP * HID) return;
    const int k = t >> 6;
    Wp[t] = (k < K) ? W[t] : 0.0f;
}

// ------------------------- WMMA GEMM:  H = XS @ W --------------------------
// XS: [n_rows, KP] prescaled+zero-padded, W: [KP, 64], H: [n_rows, 64].
// One wave32 per 16x16 output tile; V_WMMA_F32_16X16X4_F32, KP/4 steps,
// fully unrolled, branch-free inner loop.  W staged through LDS and shared
// by all 8 waves of the block.
//
// A layout (ISA 7.12.2): lanes 0-15 hold M=lane, K={k0,k0+1}; lanes 16-31
// hold M=lane-16, K={k0+2,k0+3}.  B mirrors with N across lanes.
// C/D layout: VGPR i -> M=i (lanes 0-15) / M=8+i (lanes 16-31), N = lane%16.

template <int KP>
__global__ void gcn_gemm_wmma(const float* __restrict__ XS,
                              const float* __restrict__ W,
                              float* __restrict__ H,
                              int n_rows, int totalTiles) {
    __shared__ float Ws[KP * HID];

    // cooperative stage of W into LDS (float4 per thread-iteration)
    for (int i = threadIdx.x; i < (KP * HID) / 4; i += blockDim.x)
        ((v4f*)Ws)[i] = ((const v4f*)W)[i];
    __syncthreads();

    const int lane = threadIdx.x & 31;
    const int wave = threadIdx.x >> 5;
    const int tile = blockIdx.x * (blockDim.x >> 5) + wave;
    if (tile < totalTiles) {                        // wave-uniform
        const int colTile = tile & 3;               // HID/16 == 4
        const int rowTile = tile >> 2;

        const int half = lane >> 4;                 // 0 -> K {k0,k0+1}; 1 -> {k0+2,k0+3}
        const int lm   = lane & 15;
        int arow = rowTile * 16 + lm;
        if (arow >= n_rows) arow = n_rows - 1;      // branch-free clamp
        const int col = colTile * 16 + lm;

        const float* __restrict__ xrow = XS + (long long)arow * KP;
        const float* __restrict__ wcol = Ws + col;

        v8f acc = {};
        #pragma unroll
        for (int s = 0; s < KP / 4; ++s) {
            const int k0 = s * 4 + half * 2;        // even -> 8B-aligned v2f load
            v2f a = *(const v2f*)(xrow + k0);
            v2f b;
            b.x = wcol[k0 * HID];
            b.y = wcol[(k0 + 1) * HID];
            acc = __builtin_amdgcn_wmma_f32_16x16x4_f32(
                /*neg_a=*/false, a, /*neg_b=*/false, b,
                /*c_mod=*/(short)0, acc, /*reuse_a=*/false, /*reuse_b=*/false);
        }

        // Epilogue: wave-uniform full-tile fast path -> unguarded store clause.
        const int rbase = rowTile * 16 + half * 8;
        float* __restrict__ hp = H + (long long)rbase * HID + col;
        if (rowTile * 16 + 16 <= n_rows) {          // uniform branch (always, N%16==0)
            #pragma unroll
            for (int i = 0; i < 8; ++i)
                hp[(long long)i * HID] = acc[i];
        } else {
            #pragma unroll
            for (int i = 0; i < 8; ++i)
                if (rbase + i < n_rows) hp[(long long)i * HID] = acc[i];
        }
    }
}

// ----------------------- edge scatter:  agg[dst] += h[src] -----------------
// Thread t -> (edge e = t/16, quad q = t%16): one b128 gather + 4 f32 atomics.

__global__ void scatter_edges(const float* __restrict__ H, const int* __restrict__ src,
                              const int* __restrict__ dst, float* __restrict__ AGG,
                              int n_edges) {
    long long t = (long long)blockIdx.x * blockDim.x + threadIdx.x;
    if (t >= (long long)n_edges * (HID / 4)) return;
    const int e = (int)(t >> 4);
    const int q = (int)(t & 15);
    const v4f v = *(const v4f*)(H + (long long)src[e] * HID + q * 4);
    float* dp = AGG + (long long)dst[e] * HID + q * 4;
    atomicAdd(dp + 0, v.x);
    atomicAdd(dp + 1, v.y);
    atomicAdd(dp + 2, v.z);
    atomicAdd(dp + 3, v.w);
}

// --------------------- x = relu(agg * norm_dst + bias) ---------------------

__global__ void bias_relu_norm(const float* __restrict__ AGG,
                               const float* __restrict__ norm_dst,
                               const float* __restrict__ bias,
                               float* __restrict__ Xout, int n_rows) {
    long long t = (long long)blockIdx.x * blockDim.x + threadIdx.x;
    if (t >= (long long)n_rows * (HID / 4)) return;
    const int r = (int)(t >> 4);
    const int q = (int)(t & 15);
    const float nd = norm_dst[r];
    const v4f a = *(const v4f*)(AGG + (long long)r * HID + q * 4);
    const v4f b = *(const v4f*)(bias + q * 4);
    v4f o;
    o.x = fmaxf(a.x * nd + b.x, 0.0f);
    o.y = fmaxf(a.y * nd + b.y, 0.0f);
    o.z = fmaxf(a.z * nd + b.z, 0.0f);
    o.w = fmaxf(a.w * nd + b.w, 0.0f);
    *(v4f*)(Xout + (long long)r * HID + q * 4) = o;
}

// ------------------------- column mean -> d_out[64] ------------------------

__global__ void mean_reduce(const float* __restrict__ X, float* __restrict__ out,
                            int n_rows) {
    const int d = threadIdx.x;                      // blockDim.x == 64
    float s = 0.0f;
    for (int r = blockIdx.x; r < n_rows; r += gridDim.x)
        s += X[(long long)r * HID + d];
    atomicAdd(&out[d], s / (float)n_rows);
}

// ------------------------------- launcher ----------------------------------

extern "C" void kernel_launch(void* const* d_in, const int* in_sizes, int n_in,
                              void* d_out, int out_size, void* d_ws, size_t ws_size,
                              hipStream_t stream) {
    const float* features = (const float*)d_in[0];
    const int*   src      = (const int*)d_in[1];
    const int*   dst      = (const int*)d_in[2];
    const float* W1 = (const float*)d_in[3];
    const float* b1 = (const float*)d_in[4];
    const float* W2 = (const float*)d_in[5];
    const float* b2 = (const float*)d_in[6];
    const float* W3 = (const float*)d_in[7];
    const float* b3 = (const float*)d_in[8];

    const int N = in_sizes[0] / IN_DIM;   // 50000
    const int E = in_sizes[1];            // 800000
    const int KP1 = 32;                   // padded K for layer 1

    // workspace layout (floats)
    float* ws       = (float*)d_ws;
    float* norm_src = ws;                               // [N]
    float* norm_dst = norm_src + N;                     // [N]
    float* xbuf     = norm_dst + N;                     // [N,64]
    float* hbuf     = xbuf + (size_t)N * HID;           // [N,64]
    float* agg      = hbuf + (size_t)N * HID;           // [N,64]
    float* xs       = agg  + (size_t)N * HID;           // [N,64] (prescaled A)
    float* wp       = xs   + (size_t)N * HID;           // [32,64] padded W1

    float* outp = (float*)d_out;

    const long long nh  = (long long)N * HID;
    const long long nh4 = (long long)N * (HID / 4);
    const long long eh4 = (long long)E * (HID / 4);
    const int ZB = 256;

    // degree norms
    fill_zero_f32<<<(int)((2LL * N + ZB - 1) / ZB), ZB, 0, stream>>>(norm_src, 2LL * N);
    degree_accum<<<(E + ZB - 1) / ZB, ZB, 0, stream>>>(src, dst, norm_src, norm_dst, E);
    norm_finalize<<<(N + ZB - 1) / ZB, ZB, 0, stream>>>(norm_src, norm_dst, N);

    const int rowTiles   = (N + 15) >> 4;
    const int totalTiles = rowTiles * (HID / 16);
    const int gemmBlk    = 256;                          // 8 waves / block
    const int gemmGrid   = (totalTiles + 7) / 8;
    const int nhGrid     = (int)((nh + ZB - 1) / ZB);
    const int nh4Grid    = (int)((nh4 + ZB - 1) / ZB);
    const int eh4Grid    = (int)((eh4 + ZB - 1) / ZB);
    const long long nk1  = (long long)N * KP1;
    const int nk1Grid    = (int)((nk1 + ZB - 1) / ZB);

    // ---- layer 1: features [N,23], W1 padded to [32,64] ----
    pad_weight<<<(KP1 * HID + ZB - 1) / ZB, ZB, 0, stream>>>(W1, IN_DIM, wp, KP1);
    prescale_pad<<<nk1Grid, ZB, 0, stream>>>(features, IN_DIM, IN_DIM, norm_src, xs, KP1, N);
    gcn_gemm_wmma<KP1><<<gemmGrid, gemmBlk, 0, stream>>>(xs, wp, hbuf, N, totalTiles);
    fill_zero_f32<<<nhGrid, ZB, 0, stream>>>(agg, nh);
    scatter_edges<<<eh4Grid, ZB, 0, stream>>>(hbuf, src, dst, agg, E);
    bias_relu_norm<<<nh4Grid, ZB, 0, stream>>>(agg, norm_dst, b1, xbuf, N);

    // ---- layer 2 ----
    prescale_pad<<<nhGrid, ZB, 0, stream>>>(xbuf, HID, HID, norm_src, xs, HID, N);
    gcn_gemm_wmma<HID><<<gemmGrid, gemmBlk, 0, stream>>>(xs, W2, hbuf, N, totalTiles);
    fill_zero_f32<<<nhGrid, ZB, 0, stream>>>(agg, nh);
    scatter_edges<<<eh4Grid, ZB, 0, stream>>>(hbuf, src, dst, agg, E);
    bias_relu_norm<<<nh4Grid, ZB, 0, stream>>>(agg, norm_dst, b2, xbuf, N);

    // ---- layer 3 ----
    prescale_pad<<<nhGrid, ZB, 0, stream>>>(xbuf, HID, HID, norm_src, xs, HID, N);
    gcn_gemm_wmma<HID><<<gemmGrid, gemmBlk, 0, stream>>>(xs, W3, hbuf, N, totalTiles);
    fill_zero_f32<<<nhGrid, ZB, 0, stream>>>(agg, nh);
    scatter_edges<<<eh4Grid, ZB, 0, stream>>>(hbuf, src, dst, agg, E);
    bias_relu_norm<<<nh4Grid, ZB, 0, stream>>>(agg, norm_dst, b3, xbuf, N);

    // ---- mean over nodes ----
    fill_zero_f32<<<1, HID, 0, stream>>>(outp, HID);
    mean_reduce<<<512, HID, 0, stream>>>(xbuf, outp, N);
}